// NAMTuring_55138790146747
// MI455X (gfx1250) — compile-verified
//
#include <hip/hip_runtime.h>
#include <hip/hip_bf16.h>
#include <math.h>

typedef __attribute__((ext_vector_type(2))) float v2f;
typedef __attribute__((ext_vector_type(8))) float v8f;

#define S_LEN   256
#define BATCH   64
#define INDIM   128
#define HIDDEN  128
#define NTAPE   2
#define TAPEL   128
#define DIMC    64
#define LH      64
#define GATES   256   // 4*LH

__device__ __forceinline__ float sigf(float x) { return 1.0f / (1.0f + __expf(-x)); }

// --- CDNA5 async global->LDS copies (ASYNCcnt-tracked background DMA) -------
// LDS dest = low 32 bits of the flat shared pointer (LDS aperture: flat[31:0]).
__device__ __forceinline__ void async_b128(void* lds, const void* g) {
    asm volatile("global_load_async_to_lds_b128 %0, %1, off"
                 :: "v"((uint32_t)(uintptr_t)lds), "v"(g) : "memory");
}
__device__ __forceinline__ void async_b32(void* lds, const void* g) {
    asm volatile("global_load_async_to_lds_b32 %0, %1, off"
                 :: "v"((uint32_t)(uintptr_t)lds), "v"(g) : "memory");
}
__device__ __forceinline__ void wait_async0() {
    asm volatile("s_wait_asynccnt 0" ::: "memory");
}

// ---------------------------------------------------------------------------
// C[M,N] = X[M,K] @ W[N,K]^T + b0[N] (+ b1[N])   — fp32 WMMA 16x16x4
// One 16x16 C tile per wave32; grid covers (M/16)*(N/16) tiles, 8 waves/block.
// ---------------------------------------------------------------------------
__global__ __launch_bounds__(256)
void gemm_bias_wmma(const float* __restrict__ X, const float* __restrict__ W,
                    const float* __restrict__ b0, const float* __restrict__ b1,
                    float* __restrict__ C, int M, int N, int K) {
    const int lane = threadIdx.x & 31;
    const int wave = blockIdx.x * 8 + (threadIdx.x >> 5);
    const int ntn  = N >> 4;
    const int total = (M >> 4) * ntn;
    if (wave >= total) return;               // wave-uniform: EXEC stays all-1s
    const int mt = wave / ntn, nt = wave % ntn;
    const int m0 = mt << 4, n0 = nt << 4;
    const int half = lane >> 4, l16 = lane & 15;

    const float* xrow = X + (size_t)(m0 + l16) * K;
    const float* wrow = W + (size_t)(n0 + l16) * K;
    v8f acc = {};
    for (int k0 = 0; k0 < K; k0 += 4) {
        v2f a, b;
        a.x = xrow[k0 + 2 * half + 0];
        a.y = xrow[k0 + 2 * half + 1];
        b.x = wrow[k0 + 2 * half + 0];
        b.y = wrow[k0 + 2 * half + 1];
        acc = __builtin_amdgcn_wmma_f32_16x16x4_f32(false, a, false, b,
                                                    (short)0, acc, false, false);
    }
    float bias = b0[n0 + l16];
    if (b1) bias += b1[n0 + l16];
#pragma unroll
    for (int i = 0; i < 8; ++i) {
        int m = m0 + i + 8 * half;
        C[(size_t)m * N + n0 + l16] = acc[i] + bias;
    }
}

// ---------------------------------------------------------------------------
// Persistent bidirectional LSTM scan. blockIdx.x = direction (0 fwd, 1 rev).
// LDS: h[64x64], c[64x64], g[64x256], gates_x double buffer 2x[64x256]
// (224KB dynamic of the 320KB/WGP pool). Per step: async-DMA next step's
// gates_x tile into LDS while computing G = gates_x[t] + h @ Whh^T with
// 64 WMMA tiles (8 per wave), then gate nonlinearities update c,h.
// ---------------------------------------------------------------------------
__global__ __launch_bounds__(256)
void lstm_scan(const float* __restrict__ gates_x_f, const float* __restrict__ gates_x_r,
               const float* __restrict__ Whh_f, const float* __restrict__ Whh_r,
               float* __restrict__ hidden) {
    extern __shared__ float sm[];
    float* h_lds = sm;                  // 4096
    float* c_lds = sm + 4096;           // 4096
    float* g_lds = sm + 8192;           // 16384
    float* gxb   = sm + 24576;          // 2 x 16384 (double buffer)

    const int dir = blockIdx.x;
    const float* gx  = dir ? gates_x_r : gates_x_f;
    const float* Whh = dir ? Whh_r : Whh_f;
    const int tid  = threadIdx.x;
    const int lane = tid & 31, wv = tid >> 5;
    const int half = lane >> 4, l16 = lane & 15;

    for (int i = tid; i < BATCH * LH; i += 256) { h_lds[i] = 0.f; c_lds[i] = 0.f; }

    // prefetch step 0's gates_x tile (16K floats = 4096 b128 transfers)
    {
        const int t0 = dir ? (S_LEN - 1) : 0;
        const float* src = gx + (size_t)t0 * BATCH * GATES;
        for (int i = tid; i < BATCH * GATES / 4; i += 256)
            async_b128(gxb + i * 4, src + i * 4);
    }

    for (int step = 0; step < S_LEN; ++step) {
        const int t = dir ? (S_LEN - 1 - step) : step;
        wait_async0();
        __syncthreads();                     // gates_x[t] resident in cur buffer
        const float* gxt = gxb + (step & 1) * (BATCH * GATES);

        if (step + 1 < S_LEN) {              // overlap next step's DMA with WMMA
            const int tn = dir ? (S_LEN - 2 - step) : (step + 1);
            const float* src = gx + (size_t)tn * BATCH * GATES;
            float* dst = gxb + ((step + 1) & 1) * (BATCH * GATES);
            for (int i = tid; i < BATCH * GATES / 4; i += 256)
                async_b128(dst + i * 4, src + i * 4);
        }

#pragma unroll
        for (int i = 0; i < 8; ++i) {
            const int tile = wv * 8 + i;             // 64 tiles: 4 (batch) x 16 (gate)
            const int m0 = (tile >> 4) << 4;
            const int n0 = (tile & 15) << 4;
            const float* hrow = h_lds + (m0 + l16) * LH;
            const float* wrow = Whh + (size_t)(n0 + l16) * LH;
            v8f acc = {};
#pragma unroll
            for (int k0 = 0; k0 < LH; k0 += 4) {
                v2f a, b;
                a.x = hrow[k0 + 2 * half + 0];
                a.y = hrow[k0 + 2 * half + 1];
                b.x = wrow[k0 + 2 * half + 0];
                b.y = wrow[k0 + 2 * half + 1];
                acc = __builtin_amdgcn_wmma_f32_16x16x4_f32(false, a, false, b,
                                                            (short)0, acc, false, false);
            }
#pragma unroll
            for (int r = 0; r < 8; ++r) {
                const int m = m0 + r + 8 * half;
                g_lds[m * GATES + n0 + l16] = acc[r] + gxt[m * GATES + n0 + l16];
            }
        }
        __syncthreads();
        for (int e = tid; e < BATCH * LH; e += 256) {
            const int b = e >> 6, u = e & 63;
            const float gi = g_lds[b * GATES + u];
            const float gf = g_lds[b * GATES + 64 + u];
            const float gg = g_lds[b * GATES + 128 + u];
            const float go = g_lds[b * GATES + 192 + u];
            const float cn = sigf(gf) * c_lds[e] + sigf(gi) * tanhf(gg);
            const float hn = sigf(go) * tanhf(cn);
            c_lds[e] = cn; h_lds[e] = hn;
            hidden[(size_t)t * BATCH * HIDDEN + b * HIDDEN + dir * LH + u] = hn;
        }
        __syncthreads();
    }
}

// ---------------------------------------------------------------------------
// In-place action-probability kernel: per (s,b,tape) row of 8 logits,
// softmax over [0:3], softmax over [3:6], sigmoid over [6:8].
// ---------------------------------------------------------------------------
__global__ __launch_bounds__(256)
void act_probs(float* __restrict__ a, int n) {
    const int i = blockIdx.x * blockDim.x + threadIdx.x;
    if (i >= n) return;
    float* p = a + (size_t)i * 8;
    float m = fmaxf(p[0], fmaxf(p[1], p[2]));
    float e0 = __expf(p[0] - m), e1 = __expf(p[1] - m), e2 = __expf(p[2] - m);
    float inv = 1.f / (e0 + e1 + e2);
    p[0] = e0 * inv; p[1] = e1 * inv; p[2] = e2 * inv;
    m = fmaxf(p[3], fmaxf(p[4], p[5]));
    e0 = __expf(p[3] - m); e1 = __expf(p[4] - m); e2 = __expf(p[5] - m);
    inv = 1.f / (e0 + e1 + e2);
    p[3] = e0 * inv; p[4] = e1 * inv; p[5] = e2 * inv;
    p[6] = sigf(p[6]); p[7] = sigf(p[7]);
}

// ---------------------------------------------------------------------------
// Tape scan: one block per batch element. LDS-resident tape [128][2][64]
// + double-buffered per-step val/prob staged by async DMA. 256 steps.
// ---------------------------------------------------------------------------
__global__ __launch_bounds__(256)
void tape_scan(const float* __restrict__ values, const float* __restrict__ probs,
               float* __restrict__ read_outs,
               float* __restrict__ out_tape, float* __restrict__ out_rpos,
               float* __restrict__ out_wpos) {
    extern __shared__ float sm[];
    float* tape  = sm;                          // 16384  (l*128 + t*64 + c)
    float* rpos  = tape + TAPEL * NTAPE * DIMC; // 256    (l*2 + t)
    float* wpos  = rpos + 256;                  // 256
    float* valb  = wpos + 256;                  // 2 x 128
    float* probb = valb + 256;                  // 2 x 16
    float* oldv  = probb + 32;                  // 128

    const int b = blockIdx.x;
    const int tid = threadIdx.x;

    for (int i = tid; i < TAPEL * NTAPE * DIMC; i += 256) tape[i] = 0.f;
    {   // positions: one (l,t) per thread
        const int l = tid >> 1;
        rpos[tid] = (l == 0) ? 1.f : 0.f;
        wpos[tid] = (l == 0) ? 1.f : 0.f;
    }
    // prefetch step 0 operands (128 floats val as 32 b128, 16 floats prob as b32)
    if (tid < 32)       async_b128(valb + tid * 4, values + (size_t)b * HIDDEN + tid * 4);
    else if (tid < 48)  async_b32(probb + (tid - 32), probs + (size_t)b * 16 + (tid - 32));

    for (int s = 0; s < S_LEN; ++s) {
        wait_async0();
        __syncthreads();
        const float* val  = valb + (s & 1) * 128;
        const float* prob = probb + (s & 1) * 16;

        if (s + 1 < S_LEN) {                 // overlap next step's DMA
            const size_t r = (size_t)(s + 1) * BATCH + b;
            float* vdst = valb + ((s + 1) & 1) * 128;
            float* pdst = probb + ((s + 1) & 1) * 16;
            if (tid < 32)       async_b128(vdst + tid * 4, values + r * HIDDEN + tid * 4);
            else if (tid < 48)  async_b32(pdst + (tid - 32), probs + r * 16 + (tid - 32));
        }

        if (tid < 128) {          // oldval[t,c] = sum_l tape[l,t,c] * wpos[l,t]
            const int t = tid >> 6;
            float acc = 0.f;
            for (int l = 0; l < TAPEL; ++l) acc += tape[l * 128 + tid] * wpos[l * 2 + t];
            oldv[tid] = acc;
        }
        __syncthreads();

        for (int i = tid; i < TAPEL * 128; i += 256) {   // tape += wpos*(val-old)*rw_w
            const int l = i >> 7, j = i & 127, t = j >> 6;
            tape[i] += wpos[l * 2 + t] * (val[j] - oldv[j]) * prob[t * 8 + 7];
        }
        __syncthreads();

        // position updates into registers (reads old rpos/wpos)
        const int pl = tid >> 1, pt = tid & 1;
        const float rm1 = rpos[((pl + TAPEL - 1) & (TAPEL - 1)) * 2 + pt]; // rpos[l-1]
        const float rp1 = rpos[((pl + 1) & (TAPEL - 1)) * 2 + pt];        // rpos[l+1]
        const float nw = rp1 * prob[pt * 8 + 3] + wpos[tid] * prob[pt * 8 + 4] + rm1 * prob[pt * 8 + 5];
        const float nr = rp1 * prob[pt * 8 + 0] + rpos[tid] * prob[pt * 8 + 1] + rm1 * prob[pt * 8 + 2];

        if (tid < 128) {          // read_out with NEW tape, OLD rpos, scaled by rw_r
            const int t = tid >> 6;
            float acc = 0.f;
            for (int l = 0; l < TAPEL; ++l) acc += tape[l * 128 + tid] * rpos[l * 2 + t];
            read_outs[((size_t)s * BATCH + b) * HIDDEN + tid] = acc * prob[t * 8 + 6];
        }
        __syncthreads();
        wpos[tid] = nw; rpos[tid] = nr;
        __syncthreads();
    }

    // Emit final tape [L,B,T,C] and positions [L,B,T]
    for (int i = tid; i < TAPEL * 128; i += 256) {
        const int l = i >> 7, j = i & 127;
        out_tape[(size_t)l * BATCH * 128 + b * 128 + j] = tape[i];
    }
    {
        const int l = tid >> 1, t = tid & 1;
        out_rpos[(size_t)l * BATCH * 2 + b * 2 + t] = rpos[tid];
        out_wpos[(size_t)l * BATCH * 2 + b * 2 + t] = wpos[tid];
    }
}

// ---------------------------------------------------------------------------
extern "C" void kernel_launch(void* const* d_in, const int* in_sizes, int n_in,
                              void* d_out, int out_size, void* d_ws, size_t ws_size,
                              hipStream_t stream) {
    const float* inputs = (const float*)d_in[0];
    const float* W_ih_f = (const float*)d_in[2];
    const float* W_hh_f = (const float*)d_in[3];
    const float* b_ih_f = (const float*)d_in[4];
    const float* b_hh_f = (const float*)d_in[5];
    const float* W_ih_r = (const float*)d_in[6];
    const float* W_hh_r = (const float*)d_in[7];
    const float* b_ih_r = (const float*)d_in[8];
    const float* b_hh_r = (const float*)d_in[9];
    const float* W_act  = (const float*)d_in[10];
    const float* b_act  = (const float*)d_in[11];
    const float* W_val  = (const float*)d_in[12];
    const float* b_val  = (const float*)d_in[13];
    const float* W_out  = (const float*)d_in[14];
    const float* b_out  = (const float*)d_in[15];

    const int M = S_LEN * BATCH;   // 16384 rows

    float* ws = (float*)d_ws;
    size_t off = 0;
    float* gates_f = ws + off; off += (size_t)M * GATES;
    float* gates_r = ws + off; off += (size_t)M * GATES;
    float* values  = ws + off; off += (size_t)M * HIDDEN;
    float* hidden  = ws + off; off += (size_t)M * HIDDEN;
    float* acts    = ws + off; off += (size_t)M * 16;
    float* reads   = ws + off; off += (size_t)M * HIDDEN;

    float* out      = (float*)d_out;
    float* out_y    = out;                               // [S,B,128]
    float* out_tape = out + (size_t)M * HIDDEN;          // [128,64,2,64]
    float* out_rp   = out_tape + (size_t)TAPEL * BATCH * NTAPE * DIMC;
    float* out_wp   = out_rp + (size_t)TAPEL * BATCH * NTAPE;

    // 1) input-side GEMMs (x@Wih^T + b_ih + b_hh, values)
    {
        int tiles = (M / 16) * (GATES / 16);
        gemm_bias_wmma<<<tiles / 8, 256, 0, stream>>>(inputs, W_ih_f, b_ih_f, b_hh_f,
                                                      gates_f, M, GATES, INDIM);
        gemm_bias_wmma<<<tiles / 8, 256, 0, stream>>>(inputs, W_ih_r, b_ih_r, b_hh_r,
                                                      gates_r, M, GATES, INDIM);
    }
    {
        int tiles = (M / 16) * (HIDDEN / 16);
        gemm_bias_wmma<<<tiles / 8, 256, 0, stream>>>(inputs, W_val, b_val, nullptr,
                                                      values, M, HIDDEN, INDIM);
    }

    // 2) bidirectional LSTM recurrence (persistent, 2 blocks, 224KB LDS each)
    lstm_scan<<<2, 256, (4096 + 4096 + 16384 + 2 * 16384) * sizeof(float), stream>>>(
        gates_f, gates_r, W_hh_f, W_hh_r, hidden);

    // 3) actions GEMM + probability transforms
    {
        int tiles = (M / 16) * (16 / 16);
        gemm_bias_wmma<<<tiles / 8, 256, 0, stream>>>(hidden, W_act, b_act, nullptr,
                                                      acts, M, 16, HIDDEN);
    }
    act_probs<<<(M * NTAPE + 255) / 256, 256, 0, stream>>>(acts, M * NTAPE);

    // 4) tape scan (one block per batch element)
    tape_scan<<<BATCH, 256,
                (16384 + 256 + 256 + 256 + 32 + 128) * sizeof(float), stream>>>(
        values, acts, reads, out_tape, out_rp, out_wp);

    // 5) output projection
    {
        int tiles = (M / 16) * (HIDDEN / 16);
        gemm_bias_wmma<<<tiles / 8, 256, 0, stream>>>(reads, W_out, b_out, nullptr,
                                                      out_y, M, HIDDEN, HIDDEN);
    }
}